// CPAM_Module_6356551598580
// MI455X (gfx1250) — compile-verified
//
#include <hip/hip_runtime.h>

typedef __attribute__((ext_vector_type(2))) float v2f;
typedef __attribute__((ext_vector_type(8))) float v8f;

#define TXT   153600
#define BATCH 256
#define CCH   1024
#define NPIX  196
#define C8D   128

// --------------------------------------------------------------------------
// gfx1250 async global->LDS path (ASYNCcnt), with safe synchronous fallback.
// Probe result: builtin exists, takes (v4i global*, v4i lds*, imm, imm).
// --------------------------------------------------------------------------
#if defined(__has_builtin)
#  if __has_builtin(__builtin_amdgcn_global_load_async_to_lds_b128)
#    define HAVE_ASYNC_LDS 1
#  endif
#endif

#ifdef HAVE_ASYNC_LDS
typedef __attribute__((__vector_size__(4 * sizeof(int)))) int async_i128;
typedef __attribute__((address_space(1))) async_i128* async_gp128;
typedef __attribute__((address_space(3))) async_i128* async_lp128;
#  define ASYNC_COPY16(gsrc, ldst)                                          \
      __builtin_amdgcn_global_load_async_to_lds_b128(                       \
          (async_gp128)(gsrc), (async_lp128)(ldst), 0, 0)
#  if __has_builtin(__builtin_amdgcn_s_wait_asynccnt)
#    define WAIT_ASYNC() __builtin_amdgcn_s_wait_asynccnt(0)
#  else
#    define WAIT_ASYNC() asm volatile("s_wait_asynccnt 0x0" ::: "memory")
#  endif
#else
#  define ASYNC_COPY16(gsrc, ldst) (*(float4*)(ldst) = *(const float4*)(gsrc))
#  define WAIT_ASYNC() ((void)0)
#endif

// ---------------------------------------------------------------------------
// Kernel 1: te partial GEMM.  part[z][m][j] = sum_{k in slice z} text[m][k]*G_w[j][k]
// 256 threads (8 waves), output tile 32(M) x 128(J).
// Each wave: 2 accumulators (16x16 each) sharing the A fragment.
// Double-buffered LDS, async global->LDS when available, K chunk 32.
// ---------------------------------------------------------------------------
__global__ __launch_bounds__(256) void te_gemm_kernel(
    const float* __restrict__ text, const float* __restrict__ Gw,
    float* __restrict__ part)
{
    __shared__ float As[2][32][36];    // [buf][m][k] stride 36: 16B rows, no bank conflicts
    __shared__ float Bs[2][128][36];   // [buf][j][k]

    const int t    = threadIdx.x;
    const int lane = t & 31;
    const int w    = t >> 5;        // wave 0..7
    const int wy   = w >> 2;        // 0..1 : M sub-tile (16 rows each)
    const int wx   = w & 3;         // 0..3 : J sub-tile (32 cols each)
    const int hl   = lane >> 4;
    const int ln   = lane & 15;

    const int j0   = blockIdx.x * 128;
    const int m0   = blockIdx.y * 32;
    const int z    = blockIdx.z;
    const int kbeg = z * (TXT / 4);
    const int kend = kbeg + (TXT / 4);

    const int mi = t >> 3;          // 0..31
    const int k4 = (t & 7) * 4;     // 0,4,...,28

    const float* agp = text + (size_t)(m0 + mi) * TXT + k4;
    const float* bgp = Gw + (size_t)j0 * TXT + k4;

    v8f acc0 = {0.f, 0.f, 0.f, 0.f, 0.f, 0.f, 0.f, 0.f};
    v8f acc1 = {0.f, 0.f, 0.f, 0.f, 0.f, 0.f, 0.f, 0.f};

    auto issue = [&](int k0, int buf) {
        // A tile 32x32: 1 x b128 per thread
        ASYNC_COPY16(agp + k0, &As[buf][mi][k4]);
        // B tile 128x32: 4 x b128 per thread; B[k][j] = Gw[j][k] contiguous in k
#pragma unroll
        for (int r = 0; r < 4; ++r)
            ASYNC_COPY16(bgp + (size_t)(mi + 32 * r) * TXT + k0,
                         &Bs[buf][mi + 32 * r][k4]);
    };

    issue(kbeg, 0);
    int cur = 0;
    for (int k0 = kbeg; k0 < kend; k0 += 32) {
        WAIT_ASYNC();            // own wave's async loads landed in LDS
        __syncthreads();         // everyone's landed; prev compute done
        const int nxt = cur ^ 1;
        if (k0 + 32 < kend) issue(k0 + 32, nxt);   // prefetch next chunk

        const int mr  = wy * 16 + ln;
        const int jr0 = wx * 32 + ln;
#pragma unroll
        for (int kk = 0; kk < 8; ++kk) {
            const int kb = kk * 4 + 2 * hl;
            v2f a, b0, b1;
            a.x  = As[cur][mr][kb];        a.y  = As[cur][mr][kb + 1];
            b0.x = Bs[cur][jr0][kb];       b0.y = Bs[cur][jr0][kb + 1];
            b1.x = Bs[cur][jr0 + 16][kb];  b1.y = Bs[cur][jr0 + 16][kb + 1];
            acc0 = __builtin_amdgcn_wmma_f32_16x16x4_f32(
                false, a, false, b0, (short)0, acc0, false, false);
            acc1 = __builtin_amdgcn_wmma_f32_16x16x4_f32(
                false, a, false, b1, (short)0, acc1, false, false);
        }
        cur = nxt;
    }

    float* dst = part + (size_t)z * (BATCH * CCH);
    const int row = m0 + wy * 16;
    const int c0  = j0 + wx * 32 + ln;
#pragma unroll
    for (int v = 0; v < 8; ++v) {
        dst[(size_t)(row + v + 8 * hl) * CCH + c0]      = acc0[v];
        dst[(size_t)(row + v + 8 * hl) * CCH + c0 + 16] = acc1[v];
    }
}

// ---------------------------------------------------------------------------
// Kernel 2: te = sum_z part[z] + G_b
// ---------------------------------------------------------------------------
__global__ __launch_bounds__(256) void te_reduce_kernel(
    const float* __restrict__ part, const float* __restrict__ Gb,
    float* __restrict__ te)
{
    const int i = blockIdx.x * blockDim.x + threadIdx.x;
    if (i < BATCH * CCH) {
        float v = part[i] + part[i + BATCH * CCH] +
                  part[i + 2 * BATCH * CCH] + part[i + 3 * BATCH * CCH];
        te[i] = v + Gb[i & (CCH - 1)];
    }
}

// ---------------------------------------------------------------------------
// Kernel 3: generic small WMMA GEMM:  out[i][j] = sum_k A[i][k] * W[k*sk + j*sj]
// 128 threads (4 waves), tile 16(M) x 64(J), K chunk 32 (K%32==0).
// ---------------------------------------------------------------------------
__global__ __launch_bounds__(128) void small_gemm_kernel(
    const float* __restrict__ A, int lda, int K,
    const float* __restrict__ W, int sk, int sj,
    float* __restrict__ out, int ldo)
{
    __shared__ float As[16][36];
    __shared__ float Bs[64][36];

    const int t    = threadIdx.x;
    const int lane = t & 31;
    const int w    = t >> 5;      // 0..3
    const int hl   = lane >> 4;
    const int ln   = lane & 15;
    const int j0   = blockIdx.x * 64;
    const int m0   = blockIdx.y * 16;

    const int a_mi = t >> 3;       // 0..15
    const int a_k4 = (t & 7) * 4;

    v8f acc = {0.f, 0.f, 0.f, 0.f, 0.f, 0.f, 0.f, 0.f};

    for (int k0 = 0; k0 < K; k0 += 32) {
        *(float4*)&As[a_mi][a_k4] =
            *(const float4*)&A[(size_t)(m0 + a_mi) * lda + k0 + a_k4];
#pragma unroll
        for (int r = 0; r < 16; ++r) {
            const int e  = t + 128 * r;   // 0..2047
            const int ji = e >> 5;
            const int ki = e & 31;
            Bs[ji][ki] = W[(size_t)(k0 + ki) * sk + (size_t)(j0 + ji) * sj];
        }
        __syncthreads();

        const int jr = w * 16 + ln;
#pragma unroll
        for (int kk = 0; kk < 8; ++kk) {
            const int kb = kk * 4 + 2 * hl;
            v2f a, b;
            a.x = As[ln][kb]; a.y = As[ln][kb + 1];
            b.x = Bs[jr][kb]; b.y = Bs[jr][kb + 1];
            acc = __builtin_amdgcn_wmma_f32_16x16x4_f32(
                false, a, false, b, (short)0, acc, false, false);
        }
        __syncthreads();
    }

    const int col = j0 + w * 16 + ln;
#pragma unroll
    for (int v = 0; v < 8; ++v)
        out[(size_t)(m0 + v + 8 * hl) * ldo + col] = acc[v];
}

// ---------------------------------------------------------------------------
// Kernel 4: s[b][n] = sum_c x[b][c][n] * u[b][c] + dot(b_q, kq[b])
// ---------------------------------------------------------------------------
__global__ __launch_bounds__(224) void s_kernel(
    const float* __restrict__ x, const float* __restrict__ u,
    const float* __restrict__ bq, const float* __restrict__ kq,
    float* __restrict__ s)
{
    __shared__ float su[CCH];
    const int b = blockIdx.x, t = threadIdx.x;
    for (int i = t; i < CCH; i += 224) su[i] = u[(size_t)b * CCH + i];
    __syncthreads();
    if (t < NPIX) {
        const float* xb = x + (size_t)b * CCH * NPIX + t;
        float acc = 0.f;
#pragma unroll 8
        for (int c = 0; c < CCH; ++c) acc = fmaf(xb[(size_t)c * NPIX], su[c], acc);
        float d = 0.f;
        for (int q = 0; q < C8D; ++q) d = fmaf(bq[q], kq[(size_t)b * C8D + q], d);
        s[b * NPIX + t] = acc + d;
    }
}

// ---------------------------------------------------------------------------
// Kernel 5: a[b][m] = sum_n l[n]*softmax_n(s[b][m]*l[n])
// ---------------------------------------------------------------------------
__global__ __launch_bounds__(224) void a_kernel(
    const float* __restrict__ s, const float* __restrict__ l,
    float* __restrict__ a)
{
    __shared__ float sl[NPIX];
    const int b = blockIdx.x, t = threadIdx.x;
    for (int i = t; i < NPIX; i += 224) sl[i] = l[i];
    __syncthreads();
    if (t < NPIX) {
        const float sb = s[b * NPIX + t];
        float mx = -3.4e38f;
        for (int n = 0; n < NPIX; ++n) mx = fmaxf(mx, sb * sl[n]);
        float den = 0.f, num = 0.f;
        for (int n = 0; n < NPIX; ++n) {
            const float e = __expf(sb * sl[n] - mx);
            den += e;
            num = fmaf(sl[n], e, num);
        }
        a[b * NPIX + t] = num / den;
    }
}

// ---------------------------------------------------------------------------
// Kernel 6: out[b][c][m] = gamma*(vv[b][c]*a[b][m] + b_v[c]) + x[b][c][m]
// ---------------------------------------------------------------------------
__global__ __launch_bounds__(256) void out_kernel(
    const float* __restrict__ x, const float* __restrict__ vv,
    const float* __restrict__ a, const float* __restrict__ bv,
    const float* __restrict__ gamma, float* __restrict__ out)
{
    const long long i4 = (long long)blockIdx.x * blockDim.x + threadIdx.x;
    const long long total4 = (long long)BATCH * CCH * NPIX / 4;
    if (i4 >= total4) return;
    const long long idx = i4 * 4;
    const int b   = (int)(idx / ((long long)CCH * NPIX));
    const int rem = (int)(idx - (long long)b * CCH * NPIX);
    const int c   = rem / NPIX;
    const int m   = rem - c * NPIX;          // multiple of 4 (NPIX%4==0)
    const float4 xv = ((const float4*)x)[i4];
    const float4 av = *(const float4*)&a[b * NPIX + m];
    const float vb = vv[b * CCH + c];
    const float bb = bv[c];
    const float g  = gamma[0];
    float4 o;
    o.x = fmaf(g, fmaf(vb, av.x, bb), xv.x);
    o.y = fmaf(g, fmaf(vb, av.y, bb), xv.y);
    o.z = fmaf(g, fmaf(vb, av.z, bb), xv.z);
    o.w = fmaf(g, fmaf(vb, av.w, bb), xv.w);
    ((float4*)out)[i4] = o;
}

// ---------------------------------------------------------------------------
extern "C" void kernel_launch(void* const* d_in, const int* in_sizes, int n_in,
                              void* d_out, int out_size, void* d_ws, size_t ws_size,
                              hipStream_t stream) {
    (void)in_sizes; (void)n_in; (void)out_size; (void)ws_size;
    const float* x    = (const float*)d_in[0];
    const float* text = (const float*)d_in[1];
    const float* Gw   = (const float*)d_in[2];
    const float* Gb   = (const float*)d_in[3];
    const float* l    = (const float*)d_in[4];
    const float* Wq   = (const float*)d_in[5];
    const float* bq   = (const float*)d_in[6];
    const float* Wk   = (const float*)d_in[7];
    // d_in[8] = b_k: cancels under softmax shift invariance (unused)
    const float* Wv   = (const float*)d_in[9];
    const float* bv   = (const float*)d_in[10];
    const float* gm   = (const float*)d_in[11];
    float* out = (float*)d_out;

    float* ws   = (float*)d_ws;
    float* part = ws;                         // 4 * 256*1024
    float* te   = part + 4 * BATCH * CCH;     // 256*1024
    float* kq   = te   + BATCH * CCH;         // 256*128
    float* vv   = kq   + BATCH * C8D;         // 256*1024
    float* u    = vv   + BATCH * CCH;         // 256*1024
    float* s    = u    + BATCH * CCH;         // 256*196
    float* a    = s    + BATCH * NPIX;        // 256*196

    // 1) te partials: (256 x 153600) x (153600 x 1024), K split 4-way
    te_gemm_kernel<<<dim3(CCH / 128, BATCH / 32, 4), 256, 0, stream>>>(text, Gw, part);
    // 2) reduce + bias
    te_reduce_kernel<<<(BATCH * CCH + 255) / 256, 256, 0, stream>>>(part, Gb, te);
    // 3) kq = te @ W_k.T           (W_k[j][k]: sk=1, sj=CCH)
    small_gemm_kernel<<<dim3(C8D / 64, BATCH / 16), 128, 0, stream>>>(
        te, CCH, CCH, Wk, 1, CCH, kq, C8D);
    // 4) vv = te @ W_v.T
    small_gemm_kernel<<<dim3(CCH / 64, BATCH / 16), 128, 0, stream>>>(
        te, CCH, CCH, Wv, 1, CCH, vv, CCH);
    // 5) u = kq @ W_q              (W_q[k][j]: sk=CCH, sj=1)
    small_gemm_kernel<<<dim3(CCH / 64, BATCH / 16), 128, 0, stream>>>(
        kq, C8D, C8D, Wq, CCH, 1, u, CCH);
    // 6) s[b][n] = x[b,:,n] . u[b,:] + b_q . kq[b]
    s_kernel<<<BATCH, 224, 0, stream>>>(x, u, bq, kq, s);
    // 7) softmax moment a[b][m]
    a_kernel<<<BATCH, 224, 0, stream>>>(s, l, a);
    // 8) fused rank-1 epilogue
    const long long total4 = (long long)BATCH * CCH * NPIX / 4;
    out_kernel<<<(unsigned)((total4 + 255) / 256), 256, 0, stream>>>(
        x, vv, a, bv, gm, out);
}